// ZGate_72773925863429
// MI455X (gfx1250) — compile-verified
//
#include <hip/hip_runtime.h>
#include <hip/hip_bf16.h>

// ZGate(D=2, s=1, index=2) on a 24-qubit state with B=2 columns.
// Diagonal unitary with phase (+1, -1) selected by bit 21 of the state index.
// Pure streaming op: 512 MiB HBM traffic, ~22.5 us floor at 23.3 TB/s.
// Strategy: 128-bit coalesced loads/stores, non-temporal (single-touch data),
// sign applied as an integer XOR on the float sign bit.

typedef unsigned int u4 __attribute__((ext_vector_type(4)));

__global__ __launch_bounds__(256)
void ZGate_72773925863429_kernel(const u4* __restrict__ xre,
                                 const u4* __restrict__ xim,
                                 u4* __restrict__ out,
                                 unsigned int nvec /* = n*B/4 */) {
    unsigned int t = blockIdx.x * blockDim.x + threadIdx.x;
    if (t >= nvec) return;

    // Flat float index of this chunk: f = 4*t over (n, B=2) row-major.
    // State index i = f >> 1; qubit-2 digit = bit 21 of i = bit 20 of t,
    // constant across the 4-float chunk.
    unsigned int smask = ((t >> 20) & 1u) << 31;

    // Two 128-bit non-temporal loads (global_load_b128 ... th:NT).
    u4 r = __builtin_nontemporal_load(&xre[t]);
    u4 m = __builtin_nontemporal_load(&xim[t]);

    // Sign flip on all 8 words (exact x * (-1) for re and im, incl. -0/NaN sign).
    r ^= smask;
    m ^= smask;

    // Interleave into complex64 layout: (re, im) pairs, row-major (i, b).
    u4 lo = { r.x, m.x, r.y, m.y };   // complex elems (2t,0), (2t,1)
    u4 hi = { r.z, m.z, r.w, m.w };   // complex elems (2t+1,0), (2t+1,1)

    // Two 128-bit non-temporal stores: out floats [8t .. 8t+7].
    __builtin_nontemporal_store(lo, &out[2u * t + 0u]);
    __builtin_nontemporal_store(hi, &out[2u * t + 1u]);
}

extern "C" void kernel_launch(void* const* d_in, const int* in_sizes, int n_in,
                              void* d_out, int out_size, void* d_ws, size_t ws_size,
                              hipStream_t stream) {
    (void)n_in; (void)out_size; (void)d_ws; (void)ws_size;

    const u4* xre = (const u4*)d_in[0];   // float32 (n, B) flattened
    const u4* xim = (const u4*)d_in[1];   // float32 (n, B) flattened
    u4* out = (u4*)d_out;                 // complex64 (n, B) -> interleaved floats

    // in_sizes[0] = n*B = 2^25 floats; divisible by 4 by construction (n = 2^24).
    unsigned int nflt = (unsigned int)in_sizes[0];
    unsigned int nvec = nflt >> 2;        // 2^23 threads, one uint4 pair each

    const int block = 256;                // 8 wave32s per block
    const int grid = (int)((nvec + block - 1) / block);

    ZGate_72773925863429_kernel<<<grid, block, 0, stream>>>(xre, xim, out, nvec);
}